// OptNetEq_64381559767943
// MI455X (gfx1250) — compile-verified
//
#include <hip/hip_runtime.h>
#include <math.h>

#define NB   256
#define NF   784
#define NH   512
#define NC   10
#define NEQ  100
#define MLD  104          // padded leading dim for M / AAt
#define IPM_ITERS 25
#define Q_PEN   0.1
#define SIGMA_C 0.1
#define BIG     1e10

typedef __attribute__((ext_vector_type(2))) float v2f;
typedef __attribute__((ext_vector_type(8))) float v8f;

// ---------------------------------------------------------------------------
// Block-cooperative dense f64 Cholesky + triangular solves on an LDS matrix.
// ---------------------------------------------------------------------------
__device__ void chol_f(double* M, int tid, int nt) {
  for (int k = 0; k < NEQ; ++k) {
    if (tid == 0) M[k*MLD + k] = sqrt(M[k*MLD + k]);
    __syncthreads();
    double dk = M[k*MLD + k];
    for (int i = k + 1 + tid; i < NEQ; i += nt) M[i*MLD + k] /= dk;
    __syncthreads();
    int rem = NEQ - k - 1;
    for (int idx = tid; idx < rem*rem; idx += nt) {
      int i = k + 1 + idx / rem;
      int j = k + 1 + idx % rem;
      if (j <= i) M[i*MLD + j] -= M[i*MLD + k] * M[j*MLD + k];
    }
    __syncthreads();
  }
}

__device__ void fwd_f(const double* M, double* x, int tid, int nt) {
  for (int k = 0; k < NEQ; ++k) {
    if (tid == 0) x[k] /= M[k*MLD + k];
    __syncthreads();
    double xk = x[k];
    for (int i = k + 1 + tid; i < NEQ; i += nt) x[i] -= M[i*MLD + k] * xk;
    __syncthreads();
  }
}

__device__ void bwd_f(const double* M, double* x, int tid, int nt) {
  for (int k = NEQ - 1; k >= 0; --k) {
    if (tid == 0) x[k] /= M[k*MLD + k];
    __syncthreads();
    double xk = x[k];
    for (int i = tid; i < k; i += nt) x[i] -= M[k*MLD + i] * xk;
    __syncthreads();
  }
}

// ---------------------------------------------------------------------------
// Async global->LDS staging of one 112x64 f32 K-chunk of A32 (28.7 KB).
// One b128 async op per lane-iteration => 7 async ops per wave per chunk.
// ---------------------------------------------------------------------------
__device__ inline void stage_chunk(float* __restrict__ dst,
                                   const float* __restrict__ A32,
                                   int kc, int tid) {
  for (int k4 = tid; k4 < 112 * 16; k4 += 256) {   // 1792 float4 elements
    int row  = k4 >> 4;                            // 16 float4 per 64-col row
    int col4 = k4 & 15;
    const float* src = A32 + (size_t)row * NH + kc + (col4 << 2);
    unsigned ldsoff = (unsigned)(unsigned long long)(dst + (k4 << 2));
    asm volatile("global_load_async_to_lds_b128 %0, %1, off"
                 :: "v"(ldsoff), "v"(src) : "memory");
  }
}

__device__ inline void async_fence_barrier() {
  asm volatile("s_wait_asynccnt 0x0" ::: "memory");
  __syncthreads();
}

// ---------------------------------------------------------------------------
// Kernel 1: p = -relu(x @ W1^T + b1)  via V_WMMA_F32_16X16X4_F32.
// ---------------------------------------------------------------------------
__global__ __launch_bounds__(256) void fc1_relu_neg(
    const float* __restrict__ x, const float* __restrict__ W1,
    const float* __restrict__ b1, double* __restrict__ p)
{
  const int lane = threadIdx.x & 31;
  const int wid  = threadIdx.x >> 5;
  const int tile = blockIdx.x * 8 + wid;   // 512 tiles: 16 (batch) x 32 (hidden)
  const int tm = tile >> 5;
  const int tn = tile & 31;
  const int half = lane >> 4;
  const int l    = lane & 15;

  const v2f* xa = (const v2f*)(x  + (size_t)(tm*16 + l) * NF);
  const v2f* wb = (const v2f*)(W1 + (size_t)(tn*16 + l) * NF);

  v8f c = {};
  for (int k = 0; k < NF; k += 4) {
    int i2 = (k >> 1) + half;
    v2f a = xa[i2];
    v2f b = wb[i2];
    c = __builtin_amdgcn_wmma_f32_16x16x4_f32(false, a, false, b,
                                              (short)0, c, false, false);
  }
  const int col  = tn*16 + l;
  const float bv = b1[col];
#pragma unroll
  for (int r = 0; r < 8; ++r) {
    int m = tm*16 + r + 8*half;
    float h = c[r] + bv;
    h = h > 0.f ? h : 0.f;
    p[(size_t)m * NH + col] = -(double)h;
  }
}

// ---------------------------------------------------------------------------
// Kernel 2 (single block): z0 = A^T (A A^T)^{-1} 1 ; A32 = padded f32 copy of A.
// ---------------------------------------------------------------------------
__global__ __launch_bounds__(256) void setup_kernel(
    const double* __restrict__ A, double* __restrict__ z0,
    float* __restrict__ A32)
{
  extern __shared__ char smem[];
  double* M  = (double*)smem;                       // NEQ x MLD
  double* xv = (double*)(smem + NEQ*MLD*8);         // NEQ (+pad)
  const int tid = threadIdx.x, nt = blockDim.x;

  for (int idx = tid; idx < NEQ*NEQ; idx += nt) {
    int e = idx / NEQ, f = idx % NEQ;
    double acc = 0.0;
    for (int n = 0; n < NH; ++n) acc += A[e*NH + n] * A[f*NH + n];
    M[e*MLD + f] = acc;
  }
  for (int i = tid; i < NEQ; i += nt) xv[i] = 1.0;
  __syncthreads();

  chol_f(M, tid, nt);
  fwd_f(M, xv, tid, nt);
  bwd_f(M, xv, tid, nt);
  __syncthreads();

  for (int n = tid; n < NH; n += nt) {
    double acc = 0.0;
    for (int e = 0; e < NEQ; ++e) acc += A[e*NH + n] * xv[e];
    z0[n] = acc;
  }
  for (int idx = tid; idx < 112*NH; idx += nt) {
    int r = idx / NH;
    A32[idx] = (r < NEQ) ? (float)A[idx] : 0.0f;
  }
}

// ---------------------------------------------------------------------------
// Kernel 3: persistent IPM, one workgroup per batch element.
// ---------------------------------------------------------------------------
__global__ __launch_bounds__(256) void ipm_kernel(
    const double* __restrict__ A,    // [NEQ, NH] f64
    const float*  __restrict__ A32,  // [112, NH] f32, rows >= 100 zero
    const double* __restrict__ p_g,  // [NB, NH]
    const double* __restrict__ z0,   // [NH]
    const float*  __restrict__ W2, const float* __restrict__ b2,
    float* __restrict__ out)
{
  extern __shared__ char smem[];
  double* M   = (double*)smem;                  // 100 x 104 f64
  double* zv  = (double*)(smem + 83200);
  double* sv  = zv  + NH;
  double* lv  = sv  + NH;
  double* pv  = lv  + NH;
  double* r1v = pv  + NH;
  double* hiv = r1v + NH;
  double* dzv = hiv + NH;
  float*  sqw = (float*)(dzv + NH);             // NH f32: sqrt(1/hd)
  double* yv  = (double*)(sqw + NH);            // 128
  double* dyv = yv  + 128;
  double* rhv = dyv + 128;
  double* red = rhv + 128;                      // 256
  double* scl = red + 256;                      // 2 scalars (16B)
  float*  As0 = (float*)(scl + 2);              // 112x64 chunk buffer 0
  float*  As1 = As0 + 112*64;                   // 112x64 chunk buffer 1

  const int tid  = threadIdx.x, nt = blockDim.x;
  const int lane = tid & 31,   wid = tid >> 5;
  const int b    = blockIdx.x;
  const int half = lane >> 4,  l   = lane & 15;
  const int n0 = tid, n1 = tid + 256;

  // decode this wave's owned upper-triangle 16x16 tiles of the 7x7 tiling
  int own = 0, tiA[4], tjA[4];
  for (int t = wid; t < 28; t += 8) {
    int ti = 0, tj = 0, c2 = t;
    for (int a = 0; a < 7; ++a) {
      int row = 7 - a;
      if (c2 < row) { ti = a; tj = a + c2; break; }
      c2 -= row;
    }
    tiA[own] = ti; tjA[own] = tj; ++own;
  }

  for (int n = tid; n < NH; n += nt) {
    zv[n] = z0[n];
    pv[n] = p_g[(size_t)b * NH + n];
    sv[n] = 1.0; lv[n] = 1.0;
  }
  for (int e = tid; e < NEQ; e += nt) yv[e] = 0.0;
  __syncthreads();

  for (int it = 0; it < IPM_ITERS; ++it) {
    // -------- mu = mean(s*lam) --------
    {
      double acc = sv[n0]*lv[n0] + sv[n1]*lv[n1];
      red[tid] = acc; __syncthreads();
      for (int s = nt >> 1; s > 0; s >>= 1) {
        if (tid < s) red[tid] += red[tid + s];
        __syncthreads();
      }
      if (tid == 0) scl[0] = red[0] / (double)NH;
      __syncthreads();
    }
    const double mu = scl[0];

    // -------- residuals -> r1, 1/hd, sqrt(1/hd) --------
    {
      double ya0 = 0.0, ya1 = 0.0;
      for (int e = 0; e < NEQ; ++e) {
        double ye = yv[e];
        ya0 += ye * A[e*NH + n0];
        ya1 += ye * A[e*NH + n1];
      }
#pragma unroll
      for (int q2 = 0; q2 < 2; ++q2) {
        int n = q2 ? n1 : n0;  double ya = q2 ? ya1 : ya0;
        double z_ = zv[n], s_ = sv[n], l_ = lv[n];
        double d_ = l_ / s_;
        double ri = s_ - z_;
        double rx = Q_PEN*z_ + pv[n] + ya - l_;
        double r1 = -rx + d_*ri - l_ + SIGMA_C*mu/s_;
        double hi = 1.0 / (Q_PEN + d_);
        r1v[n] = r1; hiv[n] = hi; sqw[n] = (float)sqrt(hi);
      }
      __syncthreads();
    }

    // -------- rhs = A*(r1/hd) + (A z - 1), wave-split dots --------
    for (int e = wid; e < NEQ; e += 8) {
      double a1 = 0.0, a2 = 0.0;
      for (int n = lane; n < NH; n += 32) {
        double a = A[e*NH + n];
        a1 += a * zv[n];
        a2 += a * r1v[n] * hiv[n];
      }
      for (int off = 16; off > 0; off >>= 1) {
        a1 += __shfl_down(a1, off, 32);
        a2 += __shfl_down(a2, off, 32);
      }
      if (lane == 0) rhv[e] = a2 + (a1 - 1.0);
    }

    // -------- M = A diag(1/hd) A^T --------
    // Double-buffered async global->LDS chunk staging + f32 WMMA with
    // per-64-K promotion into f64 accumulators.
    {
      double cacc[4][8];
#pragma unroll
      for (int s = 0; s < 4; ++s)
#pragma unroll
        for (int r = 0; r < 8; ++r) cacc[s][r] = 0.0;

      stage_chunk(As0, A32, 0, tid);
      async_fence_barrier();

      for (int kc = 0; kc < NH; kc += 64) {
        const int parity = (kc >> 6) & 1;
        float* cur = parity ? As1 : As0;
        float* nxt = parity ? As0 : As1;
        if (kc + 64 < NH) stage_chunk(nxt, A32, kc + 64, tid);

        const v2f* pw = (const v2f*)(sqw + kc);
        for (int s = 0; s < own; ++s) {
          const v2f* pa = (const v2f*)(cur + (tiA[s]*16 + l) * 64);
          const v2f* pb = (const v2f*)(cur + (tjA[s]*16 + l) * 64);
          v8f c = {};
#pragma unroll
          for (int kk = 0; kk < 64; kk += 4) {
            int i2 = (kk >> 1) + half;
            v2f wv = pw[i2];
            v2f a  = pa[i2]; a.x  *= wv.x; a.y  *= wv.y;
            v2f bb = pb[i2]; bb.x *= wv.x; bb.y *= wv.y;
            c = __builtin_amdgcn_wmma_f32_16x16x4_f32(false, a, false, bb,
                                                      (short)0, c, false, false);
          }
#pragma unroll
          for (int r = 0; r < 8; ++r) cacc[s][r] += (double)c[r];
        }
        async_fence_barrier();   // next chunk landed; cur free for reuse
      }

      for (int s = 0; s < own; ++s) {
#pragma unroll
        for (int r = 0; r < 8; ++r) {
          int e = tiA[s]*16 + r + 8*half;
          int f = tjA[s]*16 + l;
          if (e < NEQ && f < NEQ) {
            M[e*MLD + f] = cacc[s][r];
            M[f*MLD + e] = cacc[s][r];
          }
        }
      }
    }
    __syncthreads();

    // -------- dy = M^{-1} rhs (Cholesky) --------
    chol_f(M, tid, nt);
    for (int e = tid; e < NEQ; e += nt) dyv[e] = rhv[e];
    __syncthreads();
    fwd_f(M, dyv, tid, nt);
    bwd_f(M, dyv, tid, nt);
    __syncthreads();

    // -------- dz, fraction-to-boundary alpha --------
    {
      double da0 = 0.0, da1 = 0.0;
      for (int e = 0; e < NEQ; ++e) {
        double de = dyv[e];
        da0 += de * A[e*NH + n0];
        da1 += de * A[e*NH + n1];
      }
      double rat = BIG;
#pragma unroll
      for (int q2 = 0; q2 < 2; ++q2) {
        int n = q2 ? n1 : n0;  double da = q2 ? da1 : da0;
        double z_ = zv[n], s_ = sv[n], l_ = lv[n];
        double d_ = l_ / s_, ri = s_ - z_;
        double dz = (r1v[n] - da) * hiv[n];
        dzv[n] = dz;
        double ds = dz - ri;
        double dl = d_*(ri - dz) - l_ + SIGMA_C*mu/s_;
        if (ds < 0.0) rat = fmin(rat, -s_/ds);
        if (dl < 0.0) rat = fmin(rat, -l_/dl);
      }
      red[tid] = rat; __syncthreads();
      for (int s = nt >> 1; s > 0; s >>= 1) {
        if (tid < s) red[tid] = fmin(red[tid], red[tid + s]);
        __syncthreads();
      }
      if (tid == 0) scl[1] = 0.99 * fmin(1.0, red[0]);
      __syncthreads();
    }
    const double alpha = scl[1];

    // -------- update state --------
#pragma unroll
    for (int q2 = 0; q2 < 2; ++q2) {
      int n = q2 ? n1 : n0;
      double z_ = zv[n], s_ = sv[n], l_ = lv[n];
      double d_ = l_ / s_, ri = s_ - z_;
      double dz = dzv[n];
      double ds = dz - ri;
      double dl = d_*(ri - dz) - l_ + SIGMA_C*mu/s_;
      zv[n] = z_ + alpha*dz;
      sv[n] = s_ + alpha*ds;
      lv[n] = l_ + alpha*dl;
    }
    for (int e = tid; e < NEQ; e += nt) yv[e] += alpha*dyv[e];
    __syncthreads();
  }

  // -------- fc2 + log_softmax --------
  for (int c = wid; c < NC; c += 8) {
    double acc = 0.0;
    for (int n = lane; n < NH; n += 32)
      acc += (double)((float)zv[n]) * (double)W2[c*NH + n];
    for (int off = 16; off > 0; off >>= 1) acc += __shfl_down(acc, off, 32);
    if (lane == 0) red[c] = acc + (double)b2[c];
  }
  __syncthreads();
  if (tid == 0) {
    double mx = red[0];
    for (int c = 1; c < NC; ++c) mx = fmax(mx, red[c]);
    double se = 0.0;
    for (int c = 0; c < NC; ++c) se += exp(red[c] - mx);
    double lse = mx + log(se);
    for (int c = 0; c < NC; ++c)
      out[(size_t)b * NC + c] = (float)(red[c] - lse);
  }
}

// ---------------------------------------------------------------------------
extern "C" void kernel_launch(void* const* d_in, const int* in_sizes, int n_in,
                              void* d_out, int out_size, void* d_ws, size_t ws_size,
                              hipStream_t stream) {
  const float*  x  = (const float*)d_in[0];
  const float*  W1 = (const float*)d_in[1];
  const float*  b1 = (const float*)d_in[2];
  const float*  W2 = (const float*)d_in[3];
  const float*  b2 = (const float*)d_in[4];
  const double* A  = (const double*)d_in[5];
  float* out = (float*)d_out;

  char* ws = (char*)d_ws;
  double* p   = (double*)ws;                              // NB*NH f64
  double* z0  = (double*)(ws + (size_t)NB*NH*8);          // NH f64
  float*  A32 = (float*) (ws + (size_t)NB*NH*8 + NH*8);   // 112*NH f32

  fc1_relu_neg<<<64, 256, 0, stream>>>(x, W1, b1, p);
  setup_kernel<<<1, 256, (size_t)(NEQ*MLD + 128)*8, stream>>>(A, z0, A32);

  // M(83200) + 7 vec(28672) + sqw(2048) + y/dy/rhs(3072) + red(2048) + scl(16)
  // + 2 async chunk buffers (2*28672) = 176400 bytes
  size_t lds3 = 83200 + 7*(size_t)NH*8 + (size_t)NH*4 + 3*128*8 + 256*8 + 16
              + 2*(size_t)112*64*4;
  ipm_kernel<<<NB, 256, lds3, stream>>>(A, A32, p, z0, W2, b2, out);
}